// Model_57071525429794
// MI455X (gfx1250) — compile-verified
//
#include <hip/hip_runtime.h>

// ---------------------------------------------------------------------------
// MI455X (gfx1250) implementation.
//   trunk convs: direct fp32 VALU kernels (~3.4 GFLOP total, L2-resident)
//   bilinear head projection (M=8640,K=2080,N=64): V_WMMA_F32_16X16X4_F32,
//   z-fragments generated on the fly via triangular *segment* iteration
//   (k -> (i,j) mapping kept in registers; no index tables, no byte unpacking).
// Workspace: 3 ping-pong buffers (26.5 MB) + zero-padded ct_w copy (0.53 MB).
// ---------------------------------------------------------------------------

#define BN_S 0.9999950000374997f  // 1/sqrt(1+1e-5)

typedef __attribute__((ext_vector_type(2))) float v2f;
typedef __attribute__((ext_vector_type(8))) float v8f;

// ---------------------------------------------------------------------------
// Elementwise / conv kernels (fp32 VALU)
// ---------------------------------------------------------------------------

__global__ void bnrelu_k(const float* __restrict__ in, const float* __restrict__ g,
                         const float* __restrict__ b, float* __restrict__ out,
                         int C, int HW, int total) {
  int idx = blockIdx.x * blockDim.x + threadIdx.x;
  if (idx >= total) return;
  int c = (idx / HW) % C;
  float v = fmaf(in[idx], g[c] * BN_S, b[c]);
  out[idx] = v > 0.0f ? v : 0.0f;
}

__global__ void add_k(const float* __restrict__ a, const float* __restrict__ b,
                      float* __restrict__ out, int total) {
  int idx = blockIdx.x * blockDim.x + threadIdx.x;
  if (idx >= total) return;
  out[idx] = a[idx] + b[idx];
}

__global__ void conv3x3_k(const float* __restrict__ in, const float* __restrict__ w,
                          float* __restrict__ out, int B, int Cin, int Hin, int Win,
                          int Cout, int Hout, int Wout, int stride) {
  int idx = blockIdx.x * blockDim.x + threadIdx.x;
  int total = B * Cout * Hout * Wout;
  if (idx >= total) return;
  int wo = idx % Wout; int t = idx / Wout;
  int ho = t % Hout;   t /= Hout;
  int co = t % Cout;   int b = t / Cout;
  int hi0 = ho * stride - 1, wi0 = wo * stride - 1;
  const float* wt = w + co * Cin * 9;
  float acc = 0.0f;
  for (int ci = 0; ci < Cin; ++ci) {
    const float* ip = in + ((size_t)(b * Cin + ci) * Hin) * Win;
    const float* wp = wt + ci * 9;
#pragma unroll
    for (int kh = 0; kh < 3; ++kh) {
      int hi = hi0 + kh;
      if ((unsigned)hi >= (unsigned)Hin) continue;
      const float* row = ip + hi * Win;
#pragma unroll
      for (int kw = 0; kw < 3; ++kw) {
        int wi = wi0 + kw;
        if ((unsigned)wi >= (unsigned)Win) continue;
        acc = fmaf(row[wi], wp[kh * 3 + kw], acc);
      }
    }
  }
  out[idx] = acc;
}

__global__ void conv1x1s2_k(const float* __restrict__ in, const float* __restrict__ w,
                            float* __restrict__ out, int B, int Cin, int Hin, int Win,
                            int Cout, int Hout, int Wout) {
  int idx = blockIdx.x * blockDim.x + threadIdx.x;
  int total = B * Cout * Hout * Wout;
  if (idx >= total) return;
  int wo = idx % Wout; int t = idx / Wout;
  int ho = t % Hout;   t /= Hout;
  int co = t % Cout;   int b = t / Cout;
  int hi = ho * 2, wi = wo * 2;
  float acc = 0.0f;
  for (int ci = 0; ci < Cin; ++ci)
    acc = fmaf(in[((size_t)(b * Cin + ci) * Hin + hi) * Win + wi], w[co * Cin + ci], acc);
  out[idx] = acc;
}

// h3 (B,64,9,15) -> xp (B*135, 64) channels-last with fused final bn+relu
__global__ void bnrelu_tr_k(const float* __restrict__ in, const float* __restrict__ g,
                            const float* __restrict__ b, float* __restrict__ xp) {
  int idx = blockIdx.x * blockDim.x + threadIdx.x;   // total = 64*135*64
  if (idx >= 64 * 135 * 64) return;
  int c = idx & 63; int p = idx >> 6;
  int w = p % 15; int h = (p / 15) % 9; int bb = p / 135;
  float v = in[((size_t)(bb * 64 + c) * 9 + h) * 15 + w];
  v = fmaf(v, g[c] * BN_S, b[c]);
  xp[idx] = v > 0.0f ? v : 0.0f;
}

// Copy ct_w into workspace with 8 zero floats of tail padding so the ragged
// last WMMA chunk of the triangular sweep can read k up to 2082 in-bounds.
__global__ void padw_k(const float* __restrict__ w, float* __restrict__ wp) {
  int idx = blockIdx.x * blockDim.x + threadIdx.x;
  if (idx >= 64 * 2080 + 8) return;
  wp[idx] = (idx < 64 * 2080) ? w[idx] : 0.0f;
}

// ---------------------------------------------------------------------------
// Bilinear projection GEMM via V_WMMA_F32_16X16X4_F32.
//   A[p,o] = sum_{i<=j} xp[p,i]*xp[p,j] * ct_w[o,k(i,j)] + ct_b[o]
// Triangular K is swept segment-wise: outer i, inner j in chunks of 4 with the
// running k offset in a register. tileX rows are zero-padded to 68 columns so
// out-of-segment A-slots contribute exactly 0 (no per-slot conditionals), and
// stride 68 makes the 16-row column reads bank-conflict-free (4m+j mod 64).
// 544 WMMAs per wave (vs 520 ideal, +4.6% padding). All branches wave-uniform:
// EXEC stays all-ones as WMMA requires.
// ---------------------------------------------------------------------------
__launch_bounds__(128)
__global__ void bilinear_gemm_k(const float* __restrict__ xp, const float* __restrict__ ctwp,
                                const float* __restrict__ ctb, float* __restrict__ Aout) {
  __shared__ float tileX[16][68];
  int tid = threadIdx.x;
  int p0 = blockIdx.x * 16;                 // 8640/16 = 540 blocks, exact
  if (tid < 64) tileX[tid >> 2][64 + (tid & 3)] = 0.0f;   // zero pad cols 64..67
  for (int i = tid; i < 16 * 64; i += 128)
    tileX[i >> 6][i & 63] = xp[(size_t)p0 * 64 + i];
  __syncthreads();

  int lane = tid & 31;
  int lane16 = lane & 15;
  int khalf = (lane >> 4) << 1;             // lanes 0-15 hold K{0,1}; 16-31 hold K{2,3}
  int o = (tid >> 5) * 16 + lane16;         // this lane's N column (B-frag / store)
  const float* wrow = ctwp + (size_t)o * 2080;
  const float* rowX = tileX[lane16];        // this lane's M row (A-frag)

  v8f acc = {};
  int koff = 0;
  for (int i = 0; i < 64; ++i) {
    int len = 64 - i;                       // segment i covers j = i..63
    float xi = rowX[i];
    const float* xj = rowX + i + khalf;     // contiguous j reads, no tables
    const float* wk = wrow + koff + khalf;  // matching contiguous k reads
    for (int jb = 0; jb < len; jb += 4) {
      v2f a, b;
      a.x = xi * xj[jb];
      a.y = xi * xj[jb + 1];
      b.x = wk[jb];
      b.y = wk[jb + 1];
      acc = __builtin_amdgcn_wmma_f32_16x16x4_f32(
          /*neg_a=*/false, a, /*neg_b=*/false, b,
          /*c_mod=*/(short)0, acc, /*reuse_a=*/false, /*reuse_b=*/false);
    }
    koff += len;
  }

  float bias = ctb[o];
  int mbase = (lane < 16) ? 0 : 8;          // C/D layout: VGPR v -> M = v (+8 high half)
#pragma unroll
  for (int v = 0; v < 8; ++v)
    Aout[(size_t)(p0 + mbase + v) * 64 + o] = acc[v] + bias;
}

// G[b,o] = mean_{h,w} A[b,h,w,o]
__global__ void gmean_k(const float* __restrict__ A, float* __restrict__ G) {
  int b = blockIdx.x;      // 64 blocks
  int o = threadIdx.x;     // 64 threads
  float s = 0.0f;
  for (int p = 0; p < 135; ++p) s += A[(size_t)(b * 135 + p) * 64 + o];
  G[b * 64 + o] = s * (1.0f / 135.0f);
}

// out = xp * (A*G + A) + xp
__global__ void final_k(const float* __restrict__ xp, const float* __restrict__ A,
                        const float* __restrict__ G, float* __restrict__ out) {
  int idx = blockIdx.x * blockDim.x + threadIdx.x;   // total 552960
  if (idx >= 64 * 135 * 64) return;
  int o = idx & 63; int p = idx >> 6; int b = p / 135;
  float a = A[idx];
  float dg = fmaf(a, G[b * 64 + o], a);
  out[idx] = fmaf(xp[idx], dg, xp[idx]);
}

// ---------------------------------------------------------------------------
// Host-side orchestration
// ---------------------------------------------------------------------------
static inline int cdiv(int a, int b) { return (a + b - 1) / b; }

extern "C" void kernel_launch(void* const* d_in, const int* in_sizes, int n_in,
                              void* d_out, int out_size, void* d_ws, size_t ws_size,
                              hipStream_t stream) {
  (void)in_sizes; (void)n_in; (void)out_size; (void)ws_size;

  const float* x      = (const float*)d_in[0];
  // d_in[1] = y : unused by the reference forward path
  const float* conv_w = (const float*)d_in[2];
  const float* s1_g1 = (const float*)d_in[3];
  const float* s1_b1 = (const float*)d_in[4];
  const float* s1_w1 = (const float*)d_in[5];
  const float* s1_g2 = (const float*)d_in[6];
  const float* s1_b2 = (const float*)d_in[7];
  const float* s1_w2 = (const float*)d_in[8];
  const float* s2_g1 = (const float*)d_in[9];
  const float* s2_b1 = (const float*)d_in[10];
  const float* s2_w1 = (const float*)d_in[11];
  const float* s2_g2 = (const float*)d_in[12];
  const float* s2_b2 = (const float*)d_in[13];
  const float* s2_w2 = (const float*)d_in[14];
  const float* s2_ws = (const float*)d_in[15];
  const float* s3_g1 = (const float*)d_in[16];
  const float* s3_b1 = (const float*)d_in[17];
  const float* s3_w1 = (const float*)d_in[18];
  const float* s3_g2 = (const float*)d_in[19];
  const float* s3_b2 = (const float*)d_in[20];
  const float* s3_w2 = (const float*)d_in[21];
  const float* s3_ws = (const float*)d_in[22];
  const float* bn_g  = (const float*)d_in[23];
  const float* bn_b  = (const float*)d_in[24];
  const float* ct_w  = (const float*)d_in[25];
  const float* ct_b  = (const float*)d_in[26];
  float* out = (float*)d_out;

  const int B = 64;
  const int N1 = B * 16 * 36 * 60;   // 2,211,840 stage-1 elements
  const int N2 = B * 32 * 18 * 30;   // 1,105,920 stage-2 elements
  const int N3 = B * 64 * 9 * 15;    //   552,960 stage-3 / head elements
  const int NW = 64 * 2080 + 8;      // padded ct_w elements

  float* bufA  = (float*)d_ws;       // three 8.85 MB ping-pong buffers
  float* bufB  = bufA + N1;
  float* bufC  = bufB + N1;
  float* ctwp  = bufC + N1;          // zero-padded ct_w copy (0.53 MB)

  const int T = 256;

  // padded ct_w staging (independent of the trunk; overlaps it on the stream queue)
  padw_k<<<cdiv(NW, T), T, 0, stream>>>(ct_w, ctwp);

  // stem: h0 = conv3x3(x, conv_w)                         -> bufA (B,16,36,60)
  conv3x3_k<<<cdiv(N1, T), T, 0, stream>>>(x, conv_w, bufA, B, 3, 36, 60, 16, 36, 60, 1);

  // ---- block 1 (stride 1, identity shortcut) ----
  bnrelu_k<<<cdiv(N1, T), T, 0, stream>>>(bufA, s1_g1, s1_b1, bufB, 16, 36 * 60, N1); // xa -> bufB
  conv3x3_k<<<cdiv(N1, T), T, 0, stream>>>(bufB, s1_w1, bufC, B, 16, 36, 60, 16, 36, 60, 1);
  bnrelu_k<<<cdiv(N1, T), T, 0, stream>>>(bufC, s1_g2, s1_b2, bufA, 16, 36 * 60, N1);
  conv3x3_k<<<cdiv(N1, T), T, 0, stream>>>(bufA, s1_w2, bufC, B, 16, 36, 60, 16, 36, 60, 1);
  add_k<<<cdiv(N1, T), T, 0, stream>>>(bufC, bufB, bufA, N1);                          // h1 -> bufA

  // ---- block 2 (stride 2, 16->32, 1x1 shortcut) ----
  bnrelu_k<<<cdiv(N1, T), T, 0, stream>>>(bufA, s2_g1, s2_b1, bufB, 16, 36 * 60, N1); // xa -> bufB
  conv3x3_k<<<cdiv(N2, T), T, 0, stream>>>(bufB, s2_w1, bufC, B, 16, 36, 60, 32, 18, 30, 2);
  bnrelu_k<<<cdiv(N2, T), T, 0, stream>>>(bufC, s2_g2, s2_b2, bufA, 32, 18 * 30, N2);
  conv3x3_k<<<cdiv(N2, T), T, 0, stream>>>(bufA, s2_w2, bufC, B, 32, 18, 30, 32, 18, 30, 1);
  conv1x1s2_k<<<cdiv(N2, T), T, 0, stream>>>(bufB, s2_ws, bufA, B, 16, 36, 60, 32, 18, 30);
  add_k<<<cdiv(N2, T), T, 0, stream>>>(bufC, bufA, bufB, N2);                          // h2 -> bufB

  // ---- block 3 (stride 2, 32->64, 1x1 shortcut) ----
  bnrelu_k<<<cdiv(N2, T), T, 0, stream>>>(bufB, s3_g1, s3_b1, bufA, 32, 18 * 30, N2); // xa -> bufA
  conv3x3_k<<<cdiv(N3, T), T, 0, stream>>>(bufA, s3_w1, bufC, B, 32, 18, 30, 64, 9, 15, 2);
  bnrelu_k<<<cdiv(N3, T), T, 0, stream>>>(bufC, s3_g2, s3_b2, bufB, 64, 9 * 15, N3);
  conv3x3_k<<<cdiv(N3, T), T, 0, stream>>>(bufB, s3_w2, bufC, B, 64, 9, 15, 64, 9, 15, 1);
  conv1x1s2_k<<<cdiv(N3, T), T, 0, stream>>>(bufA, s3_ws, bufB, B, 32, 18, 30, 64, 9, 15);
  add_k<<<cdiv(N3, T), T, 0, stream>>>(bufC, bufB, bufA, N3);                          // h3 -> bufA

  // ---- head ----
  bnrelu_tr_k<<<cdiv(N3, T), T, 0, stream>>>(bufA, bn_g, bn_b, bufC);                  // xp -> bufC
  bilinear_gemm_k<<<540, 128, 0, stream>>>(bufC, ctwp, ct_b, bufB);                    // A  -> bufB
  gmean_k<<<64, 64, 0, stream>>>(bufB, bufA);                                          // G  -> bufA
  final_k<<<cdiv(N3, T), T, 0, stream>>>(bufC, bufB, bufA, out);
}